// PerturbedTopK_67242007986757
// MI455X (gfx1250) — compile-verified
//
#include <hip/hip_runtime.h>
#include <stdint.h>

#define BB 16
#define NN 1000
#define DD 2048
#define KK 256
#define SIG 0.05f
#define TPB 256
#define ROWS_PER_BLOCK 2
#define V4 (DD / 4)        /* 512 uint4/float4 per row */
#define V4PT (V4 / TPB)    /* 2 vec4 per thread */

typedef unsigned int u32;
typedef unsigned long long u64;
typedef u32 u32x4 __attribute__((ext_vector_type(4)));
typedef u32 u32x8 __attribute__((ext_vector_type(8)));

// byte offset of a __shared__ object within the workgroup's LDS
__device__ __forceinline__ u32 lds_addr_of(const void* p) {
  return (u32)(uintptr_t)(__attribute__((address_space(3))) const void*)p;
}

// order-preserving float -> u32 key
__device__ __forceinline__ u32 f2key(u32 nbits, float xv) {
  u32 u = __float_as_uint(fmaf(__uint_as_float(nbits), SIG, xv));
  return (u & 0x80000000u) ? ~u : (u | 0x80000000u);
}

// Issue one TDM row load: 2048 x f32 tile from global to LDS (wave-uniform args)
__device__ __forceinline__ void tdm_load_row(const float* gsrc, u32 lds) {
  u64 ga = (u64)(uintptr_t)gsrc;
  u32x4 g0;
  g0[0] = 1u;                                              // count=1 valid descriptor
  g0[1] = lds;                                             // lds_addr (bytes)
  g0[2] = (u32)(ga & 0xFFFFFFFFull);                       // global_addr[31:0]
  g0[3] = (u32)((ga >> 32) & 0x01FFFFFFull) | (2u << 30);  // global_addr[56:32] | type=2
  u32x8 g1;
  g1[0] = 2u << 16;                                        // data_size = 4 bytes
  g1[1] = ((u32)DD & 0xFFFFu) << 16;                       // tensor_dim0 = 2048 (lo16)
  g1[2] = 1u << 16;                                        // tensor_dim0 hi | tensor_dim1 = 1
  g1[3] = ((u32)DD & 0xFFFFu) << 16;                       // tensor_dim1 hi | tile_dim0 = 2048
  g1[4] = 1u;                                              // tile_dim1 = 1, tile_dim2 = 0
  g1[5] = (u32)DD;                                         // tensor_dim0_stride (lo32)
  g1[6] = 0u;
  g1[7] = 0u;
  asm volatile("tensor_load_to_lds %0, %1" :: "s"(g0), "s"(g1) : "memory");
}

__global__ void zero_f32_kernel(float4* __restrict__ p, int n4) {
  int i = blockIdx.x * blockDim.x + threadIdx.x;
  int stride = gridDim.x * blockDim.x;
  const float4 z = make_float4(0.f, 0.f, 0.f, 0.f);
  for (; i < n4; i += stride) p[i] = z;
}

// Select top-K of one row held in LDS (keys buffer), scatter 1/N into outb[rank*DD + d].
__device__ void process_row(u32* keys, const float* xrow, float* outb) {
  __shared__ u32 hist[256];
  __shared__ u32 scan[256];
  __shared__ u32 skey[KK];
  __shared__ u32 sidx[KK];
  __shared__ u32 sh_prefix, sh_remaining, sh_nG;

  const u32 tid = threadIdx.x;
  uint4* k4 = (uint4*)keys;
  const float4* x4 = (const float4*)xrow;

  // closes previous phase's use of shared state AND publishes the DMA'd buffer
  __syncthreads();

  // perturbed value -> monotonic u32 key, in place (b128 LDS traffic)
  for (int j = 0; j < V4PT; ++j) {
    int i4 = (int)tid + j * TPB;
    uint4 nv = k4[i4];
    float4 xv = x4[i4];
    nv.x = f2key(nv.x, xv.x);
    nv.y = f2key(nv.y, xv.y);
    nv.z = f2key(nv.z, xv.z);
    nv.w = f2key(nv.w, xv.w);
    k4[i4] = nv;
  }
  if (tid == 0) { sh_prefix = 0u; sh_remaining = KK; sh_nG = 0u; }
  __syncthreads();

  // ---- 4 x 8-bit MSB-first radix passes: find threshold key T ----
  for (int p = 0; p < 4; ++p) {
    const int shift = 24 - 8 * p;
    hist[tid] = 0u;
    __syncthreads();
    const u32 prefix = sh_prefix;
    const u32 mask = (p == 0) ? 0u : (0xFFFFFFFFu << (shift + 8));
    for (int j = 0; j < V4PT; ++j) {
      uint4 kv = k4[tid + (u32)j * TPB];
      u32 vals[4] = {kv.x, kv.y, kv.z, kv.w};
      for (int c = 0; c < 4; ++c) {
        u32 k = vals[c];
        if (((k ^ prefix) & mask) == 0u) atomicAdd(&hist[(k >> shift) & 0xFFu], 1u);
      }
    }
    __syncthreads();
    scan[tid] = hist[tid];
    __syncthreads();
    for (int off = 1; off < 256; off <<= 1) {   // inclusive suffix sum
      u32 v = (tid + off < 256) ? scan[tid + off] : 0u;
      __syncthreads();
      scan[tid] += v;
      __syncthreads();
    }
    const u32 rem = sh_remaining;
    __syncthreads();
    const u32 sufNext = (tid < 255) ? scan[tid + 1] : 0u;
    if (scan[tid] >= rem && sufNext < rem) {    // exactly one thread fires
      sh_prefix = prefix | (tid << shift);
      sh_remaining = rem - sufNext;
    }
    __syncthreads();
  }

  const u32 T = sh_prefix;
  const u32 rem = sh_remaining;   // #elements equal to T to keep (>=1)

  // compact strictly-greater survivors (C = KK - rem <= 255)
  for (int j = 0; j < V4PT; ++j) {
    int i4 = (int)tid + j * TPB;
    uint4 kv = k4[i4];
    u32 vals[4] = {kv.x, kv.y, kv.z, kv.w};
    for (int c = 0; c < 4; ++c) {
      if (vals[c] > T) {
        u32 pos = atomicAdd(&sh_nG, 1u);
        if (pos < KK) { skey[pos] = vals[c]; sidx[pos] = (u32)(4 * i4 + c); }
      }
    }
  }
  __syncthreads();
  const u32 C = sh_nG;
  const float invN = 1.0f / (float)NN;

  // ties at T: stable ascending-index ranks C..KK-1
  for (int j = 0; j < V4PT; ++j) {
    int i4 = (int)tid + j * TPB;
    uint4 kv = k4[i4];
    u32 vals[4] = {kv.x, kv.y, kv.z, kv.w};
    for (int c = 0; c < 4; ++c) {
      if (vals[c] == T) {
        int i = 4 * i4 + c;
        u32 t = 0;
        for (int jj = 0; jj < i; ++jj) t += (keys[jj] == T) ? 1u : 0u;
        if (t < rem) atomicAdd(&outb[(u64)(C + t) * DD + (u32)i], invN);
      }
    }
  }

  // strictly-greater survivors: exact rank by counting within the set
  if (tid < C) {
    const u32 mk = skey[tid];
    const u32 mi = sidx[tid];
    u32 r = 0;
    for (u32 s = 0; s < C; ++s) {
      u32 ok = skey[s];
      r += (u32)((ok > mk) || (ok == mk && sidx[s] < mi));
    }
    atomicAdd(&outb[(u64)r * DD + mi], invN);
  }
}

__global__ __launch_bounds__(TPB) void ptopk_kernel(const float* __restrict__ x,
                                                    const float* __restrict__ noise,
                                                    float* __restrict__ out) {
  __shared__ __align__(16) u32 nb[ROWS_PER_BLOCK][DD];  // DMA dest -> keys, per row
  __shared__ __align__(16) float xrow[DD];

  const u32 tid = threadIdx.x;
  const u32 blk = blockIdx.x;
  const u32 groups_per_b = NN / ROWS_PER_BLOCK;
  const u32 b = blk / groups_per_b;
  const u32 n0 = (blk % groups_per_b) * ROWS_PER_BLOCK;

  // wave 0: issue BOTH row DMAs up front (TENSORcnt = 2, completes in order)
  if (__builtin_amdgcn_readfirstlane((int)tid) == 0) {
    const float* base = noise + ((u64)b * NN + n0) * DD;
    tdm_load_row(base, lds_addr_of(nb[0]));
    tdm_load_row(base + DD, lds_addr_of(nb[1]));
  }

  // overlap with DMA: stage the x row (shared by both rows) via b128 loads
  {
    const float4* xg = (const float4*)(x + (u64)b * DD);
    float4* xs = (float4*)xrow;
    for (int j = 0; j < V4PT; ++j) {
      int i4 = (int)tid + j * TPB;
      xs[i4] = xg[i4];
    }
  }

  float* outb = out + (u64)b * KK * DD;

  // row 0: wait until first DMA done (second may still be in flight)
  if (__builtin_amdgcn_readfirstlane((int)tid) == 0) {
    __builtin_amdgcn_s_wait_tensorcnt(1);
  }
  process_row(nb[0], xrow, outb);   // begins with __syncthreads()

  // row 1: wait for remaining DMA
  if (__builtin_amdgcn_readfirstlane((int)tid) == 0) {
    __builtin_amdgcn_s_wait_tensorcnt(0);
  }
  process_row(nb[1], xrow, outb);
}

extern "C" void kernel_launch(void* const* d_in, const int* in_sizes, int n_in,
                              void* d_out, int out_size, void* d_ws, size_t ws_size,
                              hipStream_t stream) {
  (void)in_sizes; (void)n_in; (void)d_ws; (void)ws_size;
  const float* x = (const float*)d_in[0];      // (16, 2048) f32
  const float* noise = (const float*)d_in[1];  // (16, 1000, 2048) f32
  float* out = (float*)d_out;                  // (16, 256, 2048) f32

  int n4 = out_size / 4;                       // out_size divisible by 4
  int zblocks = (n4 + TPB * 4 - 1) / (TPB * 4);
  if (zblocks < 1) zblocks = 1;
  zero_f32_kernel<<<dim3(zblocks), dim3(TPB), 0, stream>>>((float4*)out, n4);

  ptopk_kernel<<<dim3(BB * (NN / ROWS_PER_BLOCK)), dim3(TPB), 0, stream>>>(x, noise, out);
}